// MemoryModule_23244363006473
// MI455X (gfx1250) — compile-verified
//
#include <hip/hip_runtime.h>
#include <hip/hip_bf16.h>
#include <cmath>

typedef __attribute__((ext_vector_type(2))) float v2f;
typedef __attribute__((ext_vector_type(8))) float v8f;
typedef __attribute__((ext_vector_type(4))) unsigned int v4u;
typedef __attribute__((ext_vector_type(4))) int v4i;
typedef __attribute__((ext_vector_type(8))) int v8i;

__device__ __forceinline__ float wave_reduce_sum(float v) {
    #pragma unroll
    for (int off = 16; off > 0; off >>= 1)
        v += __shfl_xor(v, off, 32);
    return v;
}

// ---------------------------------------------------------------------------
// TDM: async 2D tile load global -> LDS via tensor descriptor (D#).
// tile_w elements per row (4B each), tile_h rows, row_stride elements.
// pad_interval_code/pad_amount_code add LDS padding (row stride in LDS).
// ---------------------------------------------------------------------------
__device__ __forceinline__ void tdm_load_2d(unsigned lds_off, const void* gptr,
                                            unsigned tile_w, unsigned tile_h,
                                            unsigned row_stride,
                                            unsigned pad_i, unsigned pad_a) {
    const unsigned long long ga = (unsigned long long)(size_t)gptr;
    v4u g0;
    g0.x = 1u;                                              // count=1 (valid)
    g0.y = lds_off;                                         // lds_addr
    g0.z = (unsigned)(ga & 0xFFFFFFFFull);                  // global_addr lo
    g0.w = (unsigned)((ga >> 32) & 0x01FFFFFFull) | 0x80000000u; // addr hi | type=2
    v8i g1;
    unsigned d0 = (2u << 16);                               // data_size = 4B
    d0 |= (1u << 20) | (pad_i << 22) | (pad_a << 25);       // pad_enable
    g1[0] = (int)d0;
    g1[1] = (int)((tile_w & 0xFFFFu) << 16);                // tensor_dim0 lo | bar_addr=0
    g1[2] = (int)(((tile_w >> 16) & 0xFFFFu) | ((tile_h & 0xFFFFu) << 16)); // dim0 hi | dim1 lo
    g1[3] = (int)(((tile_h >> 16) & 0xFFFFu) | (tile_w << 16));             // dim1 hi | tile_dim0
    g1[4] = (int)(tile_h & 0xFFFFu);                        // tile_dim1 | tile_dim2=0
    g1[5] = (int)row_stride;                                // tensor_dim0_stride lo
    g1[6] = 0;
    g1[7] = 0;
    const v4i z4 = {0, 0, 0, 0};
#if __clang_major__ >= 23
    const v8i z8 = {0, 0, 0, 0, 0, 0, 0, 0};
    __builtin_amdgcn_tensor_load_to_lds(g0, g1, z4, z4, z8, 0);
#else
    __builtin_amdgcn_tensor_load_to_lds(g0, g1, z4, z4, 0);
#endif
}

// ---------------------------------------------------------------------------
// Kernel A: projection GEMM.  K[b][t][h] = sum_d x[b][t][d] * W[h][d]
// TDM double-buffered LDS staging + fp32 WMMA 16x16x4.
// Block 256 thr = 8 waves (4M x 2N); block tile 64 x 128; K-step 32.
// ---------------------------------------------------------------------------
__global__ void __launch_bounds__(256)
proj_gemm(const float* __restrict__ x, const float* __restrict__ W,
          float* __restrict__ K, int T, int Lfull) {
    __shared__ float shA[2][64 * 34];    // 64 rows, LDS stride 34 (32 data + 2 pad)
    __shared__ float shB[2][128 * 34];

    const int tid  = threadIdx.x;
    const int lane = tid & 31;
    const int w    = tid >> 5;
    const int wy   = w & 3, wx = w >> 2;
    const int lo   = lane & 15, hi = lane >> 4;
    const int b    = blockIdx.z;
    const int m0b  = blockIdx.y * 64;
    const int n0b  = blockIdx.x * 128;

    const float* Ablk = x + ((size_t)b * Lfull + m0b) * 1024;
    const float* Bblk = W + (size_t)n0b * 1024;

    v8f acc0 = {}, acc1 = {}, acc2 = {}, acc3 = {};

    if (w == 0) {   // prologue: fill buffer 0
        tdm_load_2d((unsigned)(size_t)(void*)&shA[0][0], Ablk, 32, 64, 1024, 4, 1);
        tdm_load_2d((unsigned)(size_t)(void*)&shB[0][0], Bblk, 32, 128, 1024, 4, 1);
    }

    for (int kb = 0; kb < 1024; kb += 32) {
        const int buf = (kb >> 5) & 1;
        if (w == 0) {
            if (kb + 32 < 1024) {
                tdm_load_2d((unsigned)(size_t)(void*)&shA[buf ^ 1][0], Ablk + kb + 32, 32, 64, 1024, 4, 1);
                tdm_load_2d((unsigned)(size_t)(void*)&shB[buf ^ 1][0], Bblk + kb + 32, 32, 128, 1024, 4, 1);
                __builtin_amdgcn_s_wait_tensorcnt((short)2);  // current buf done
            } else {
                __builtin_amdgcn_s_wait_tensorcnt((short)0);
            }
        }
        __syncthreads();

        const float* A  = &shA[buf][0] + (wy * 16 + lo) * 34;
        const float* B0 = &shB[buf][0] + (wx * 64 +  0 + lo) * 34;
        const float* B1 = &shB[buf][0] + (wx * 64 + 16 + lo) * 34;
        const float* B2 = &shB[buf][0] + (wx * 64 + 32 + lo) * 34;
        const float* B3 = &shB[buf][0] + (wx * 64 + 48 + lo) * 34;

        #pragma unroll
        for (int kk = 0; kk < 32; kk += 4) {
            const int ko = kk + 2 * hi;
            v2f a  = *(const v2f*)(A  + ko);
            v2f b0 = *(const v2f*)(B0 + ko);
            v2f b1 = *(const v2f*)(B1 + ko);
            v2f b2 = *(const v2f*)(B2 + ko);
            v2f b3 = *(const v2f*)(B3 + ko);
            acc0 = __builtin_amdgcn_wmma_f32_16x16x4_f32(false, a, false, b0, (short)0, acc0, false, false);
            acc1 = __builtin_amdgcn_wmma_f32_16x16x4_f32(false, a, false, b1, (short)0, acc1, false, false);
            acc2 = __builtin_amdgcn_wmma_f32_16x16x4_f32(false, a, false, b2, (short)0, acc2, false, false);
            acc3 = __builtin_amdgcn_wmma_f32_16x16x4_f32(false, a, false, b3, (short)0, acc3, false, false);
        }
        __syncthreads();   // everyone done reading buf before it is re-filled
    }

    const int m0 = m0b + wy * 16;
    const int n0 = n0b + wx * 64;
    float* Kb = K + (size_t)b * T * 512;
    #pragma unroll
    for (int v = 0; v < 8; ++v) {
        const int orow = m0 + v + 8 * hi;
        if (orow < T) {
            float* dst = Kb + (size_t)orow * 512 + n0 + lo;
            dst[ 0] = acc0[v];
            dst[16] = acc1[v];
            dst[32] = acc2[v];
            dst[48] = acc3[v];
        }
    }
}

// ---------------------------------------------------------------------------
// Kernel B: chunked delta-rule scan.  One block (1024 thr) per batch.
// Per chunk of 16 steps:  U = M @ Kn^T (WMMA),  G = Kn Kn^T (WMMA),
// short sequential gate recurrence, then M += (w*D)^T @ Kn (WMMA, K=16).
// ---------------------------------------------------------------------------
#define CHUNK 16

__global__ void __launch_bounds__(1024)
scan_kernel(const float* __restrict__ KS, const float* __restrict__ KE,
            float* __restrict__ Msem, float* __restrict__ Mepi,
            float scale, int T, int Lfull) {
    extern __shared__ float smem[];
    float* sh_k   = smem;                    // [2][16][512]
    float* sh_kn  = sh_k  + 2 * 16 * 512;    // [2][16][516]
    float* sh_ud  = sh_kn + 2 * 16 * 516;    // [2][16][516]  U then D in-place
    float* sh_G   = sh_ud + 2 * 16 * 516;    // [2][16][16]
    float* sh_n2  = sh_G  + 2 * 256;         // [2][16]
    float* sh_w   = sh_n2 + 32;              // [2][16]
    float* sh_red = sh_w  + 32;              // [32]

    const int b    = blockIdx.x;
    const int tid  = threadIdx.x;
    const int lane = tid & 31;
    const int w    = tid >> 5;               // 0..31
    const int lo   = lane & 15, hi = lane >> 4;

    float* Ms = Msem + (size_t)b * 512 * 512;
    float* Me = Mepi + (size_t)b * 512 * 512;

    for (int e = tid; e < 512 * 512; e += 1024) { Ms[e] = 0.f; Me[e] = 0.f; }
    __syncthreads();

    const int mat_i = tid >> 9;              // phase-3 ownership
    const int i_i   = tid & 511;

    for (int t0 = 0; t0 < T; t0 += CHUNK) {
        const int tcount = (T - t0 < CHUNK) ? (T - t0) : CHUNK;

        // ---- phase 1: load chunk of keys --------------------------------
        for (int idx = tid; idx < 2 * CHUNK * 512; idx += 1024) {
            const int m = idx >> 13;
            const int t = (idx >> 9) & 15;
            const int i = idx & 511;
            float v = 0.f;
            if (t < tcount) {
                const float* src = m ? KE : KS;
                v = src[((size_t)b * T + (t0 + t)) * 512 + i];
            }
            sh_k[idx] = v;
        }
        __syncthreads();

        // ---- phase 1b: norms + normalized keys (wave -> (mat,t)) --------
        {
            const int m = w >> 4, t = w & 15;
            const float* kp = sh_k + (m * 16 + t) * 512;
            float s = 0.f;
            #pragma unroll
            for (int jj = 0; jj < 16; ++jj) { const float v = kp[lane + (jj << 5)]; s += v * v; }
            s = wave_reduce_sum(s);
            const float inv = 1.0f / fmaxf(sqrtf(s), 1e-12f);
            float* knp = sh_kn + (m * 16 + t) * 516;
            #pragma unroll
            for (int jj = 0; jj < 16; ++jj) { const int j = lane + (jj << 5); knp[j] = kp[j] * inv; }
            if (lane == 0) sh_n2[m * 16 + t] = s;
        }
        __syncthreads();

        // ---- phase 2: U = M @ Kn^T (2 tiles/wave) + Gram (waves 0,1) ----
        for (int f = 2 * w; f < 2 * w + 2; ++f) {
            const int m  = f >> 5;
            const int i0 = (f & 31) * 16;
            const float* Mb   = m ? Me : Ms;
            const float* arow = Mb + (size_t)(i0 + lo) * 512;
            const float* brow = sh_kn + m * 16 * 516 + lo * 516;
            v8f acc = {};
            for (int kb = 0; kb < 512; kb += 4) {
                const int ko = kb + 2 * hi;
                v2f a  = *(const v2f*)(arow + ko);
                v2f bb = *(const v2f*)(brow + ko);
                acc = __builtin_amdgcn_wmma_f32_16x16x4_f32(false, a, false, bb, (short)0, acc, false, false);
            }
            float* up = sh_ud + m * 16 * 516;
            #pragma unroll
            for (int v = 0; v < 8; ++v)
                up[lo * 516 + i0 + v + 8 * hi] = acc[v];
        }
        if (w < 2) {    // Gram: G = Kn @ Kn^T  (A frag == B frag)
            const int m = w;
            const float* r = sh_kn + m * 16 * 516 + lo * 516;
            v8f acc = {};
            for (int kb = 0; kb < 512; kb += 4) {
                const int ko = kb + 2 * hi;
                v2f a = *(const v2f*)(r + ko);
                acc = __builtin_amdgcn_wmma_f32_16x16x4_f32(false, a, false, a, (short)0, acc, false, false);
            }
            float* gp = sh_G + m * 256;
            #pragma unroll
            for (int v = 0; v < 8; ++v)
                gp[(v + 8 * hi) * 16 + lo] = acc[v];
        }
        __syncthreads();

        // ---- phase 3: sequential gate recurrence (thread owns (mat,i)) --
        bool anyF = false;
        {
            const int m = mat_i, i = i_i;
            const float* kmb = sh_k + m * 16 * 512;
            float*       dmb = sh_ud + m * 16 * 516;
            const float* gmb = sh_G + m * 256;
            const float* wmb = sh_w + m * 16;
            for (int t = 0; t < CHUNK; ++t) {
                float acc = kmb[t * 512 + i] - dmb[t * 516 + i];   // k - U[:,t]
                for (int s = 0; s < t; ++s)
                    acc -= wmb[s] * gmb[s * 16 + t] * dmb[s * 516 + i];
                dmb[t * 516 + i] = acc;                            // D[t][i]
                const float p = wave_reduce_sum(acc * acc);
                if (lane == 0) sh_red[w] = p;
                __syncthreads();
                float e2s = 0.f, e2e = 0.f;
                #pragma unroll
                for (int r2 = 0; r2 < 16; ++r2) { e2s += sh_red[r2]; e2e += sh_red[16 + r2]; }
                const bool fire = (t < tcount) &&
                    ((e2s >= 0.49f * sh_n2[t]) || (e2e >= 0.49f * sh_n2[16 + t]));
                anyF = anyF || fire;
                const float wt = (float)(t0 + t + 1) / (float)Lfull;
                const float wc = fire ? (m ? scale * wt : scale) : 0.f;
                if (i == 0) sh_w[m * 16 + t] = wc;
                __syncthreads();
            }
        }

        // ---- phase 4: M += (w*D)^T @ Kn  (rank-16 WMMA update) ----------
        if (anyF) {
            for (int f = w; f < 2048; f += 32) {
                const int m  = f >> 10;
                const int i0 = ((f >> 5) & 31) * 16;
                const int j0 = (f & 31) * 16;
                float* Mb = m ? Me : Ms;
                const float* dmb = sh_ud + m * 16 * 516;
                const float* knb = sh_kn + m * 16 * 516;
                const float* wv  = sh_w + m * 16;
                float* cp = Mb + (size_t)i0 * 512 + j0;
                v8f c;
                #pragma unroll
                for (int v = 0; v < 8; ++v)
                    c[v] = cp[(size_t)(v + 8 * hi) * 512 + lo];
                #pragma unroll
                for (int kb = 0; kb < 16; kb += 4) {
                    const int tA = kb + 2 * hi;
                    v2f a, bb;
                    a.x  = wv[tA]     * dmb[tA * 516 + i0 + lo];
                    a.y  = wv[tA + 1] * dmb[(tA + 1) * 516 + i0 + lo];
                    bb.x = knb[tA * 516 + j0 + lo];
                    bb.y = knb[(tA + 1) * 516 + j0 + lo];
                    c = __builtin_amdgcn_wmma_f32_16x16x4_f32(false, a, false, bb, (short)0, c, false, false);
                }
                #pragma unroll
                for (int v = 0; v < 8; ++v)
                    cp[(size_t)(v + 8 * hi) * 512 + lo] = c[v];
            }
        }
        __syncthreads();
    }
}

// ---------------------------------------------------------------------------
// Kernel C1: query projection, normalize, readout r = M @ qn, null gate.
// ---------------------------------------------------------------------------
__global__ void __launch_bounds__(512)
finalize_qr(const float* __restrict__ x,
            const float* __restrict__ W_sem, const float* __restrict__ W_epi,
            const float* __restrict__ null_w, const float* __restrict__ null_b,
            const float* __restrict__ Msem, const float* __restrict__ Mepi,
            float* __restrict__ rg) {
    const int b    = blockIdx.x;
    const int tid  = threadIdx.x;
    const int lane = tid & 31;
    const int w    = tid >> 5;

    __shared__ float sh_q[1024];
    __shared__ float sh_qn[1024];
    __shared__ float sh_red[16];

    const float* q = x + ((size_t)b * 2048 + 2047) * 1024;
    sh_q[tid]       = q[tid];
    sh_q[tid + 512] = q[tid + 512];

    float gp = q[tid] * null_w[tid] + q[tid + 512] * null_w[tid + 512];
    gp = wave_reduce_sum(gp);
    if (lane == 0) sh_red[w] = gp;
    __syncthreads();
    float gdot = 0.f;
    #pragma unroll
    for (int i = 0; i < 16; ++i) gdot += sh_red[i];
    __syncthreads();

    float s = 0.f, e = 0.f;
    const float* wsr = W_sem + (size_t)tid * 1024;
    const float* wer = W_epi + (size_t)tid * 1024;
    for (int d = 0; d < 1024; ++d) {
        const float qd = sh_q[d];
        s += wsr[d] * qd;
        e += wer[d] * qd;
    }
    float ps = wave_reduce_sum(s * s);
    if (lane == 0) sh_red[w] = ps;
    __syncthreads();
    float n2s = 0.f;
    #pragma unroll
    for (int i = 0; i < 16; ++i) n2s += sh_red[i];
    __syncthreads();
    float pe = wave_reduce_sum(e * e);
    if (lane == 0) sh_red[w] = pe;
    __syncthreads();
    float n2e = 0.f;
    #pragma unroll
    for (int i = 0; i < 16; ++i) n2e += sh_red[i];

    sh_qn[tid]       = s * (1.0f / fmaxf(sqrtf(n2s), 1e-12f));
    sh_qn[512 + tid] = e * (1.0f / fmaxf(sqrtf(n2e), 1e-12f));
    __syncthreads();

    const float g = 1.0f / (1.0f + expf(-(gdot + null_b[0])));
    const float* rows = Msem + ((size_t)b * 512 + tid) * 512;
    const float* rowe = Mepi + ((size_t)b * 512 + tid) * 512;
    float rs = 0.f, re = 0.f;
    for (int j = 0; j < 512; ++j) {
        rs += rows[j] * sh_qn[j];
        re += rowe[j] * sh_qn[512 + j];
    }
    rg[(size_t)b * 1024 + tid]       = g * rs;
    rg[(size_t)b * 1024 + 512 + tid] = g * re;
}

// ---------------------------------------------------------------------------
// Kernel C2: out = rg @ out_w^T + out_b  (fp32 WMMA, M=16 N=1024 K=1024)
// ---------------------------------------------------------------------------
__global__ void __launch_bounds__(256)
out_gemm(const float* __restrict__ rg, const float* __restrict__ out_w,
         const float* __restrict__ out_b, float* __restrict__ out) {
    const int tid  = threadIdx.x;
    const int lane = tid & 31;
    const int w    = tid >> 5;
    const int lo   = lane & 15, hi = lane >> 4;
    const int n0   = blockIdx.x * 128 + w * 16;

    const float* arow = rg + (size_t)lo * 1024;
    const float* brow = out_w + (size_t)(n0 + lo) * 1024;

    v8f acc = {};
    for (int kb = 0; kb < 1024; kb += 4) {
        const int ko = kb + 2 * hi;
        v2f a  = *(const v2f*)(arow + ko);
        v2f bb = *(const v2f*)(brow + ko);
        acc = __builtin_amdgcn_wmma_f32_16x16x4_f32(false, a, false, bb, (short)0, acc, false, false);
    }
    const float bias = out_b[n0 + lo];
    #pragma unroll
    for (int v = 0; v < 8; ++v)
        out[(size_t)(v + 8 * hi) * 1024 + n0 + lo] = acc[v] + bias;
}

// ---------------------------------------------------------------------------
extern "C" void kernel_launch(void* const* d_in, const int* in_sizes, int n_in,
                              void* d_out, int out_size, void* d_ws, size_t ws_size,
                              hipStream_t stream) {
    (void)in_sizes; (void)n_in; (void)out_size; (void)ws_size;
    const float* x      = (const float*)d_in[0];
    const float* W_sem  = (const float*)d_in[1];
    const float* W_epi  = (const float*)d_in[2];
    const float* null_w = (const float*)d_in[3];
    const float* null_b = (const float*)d_in[4];
    const float* out_w  = (const float*)d_in[5];
    const float* out_b  = (const float*)d_in[6];
    float* out = (float*)d_out;

    constexpr int B = 16, L = 2048, H = 512, T = L - 1;

    float* KS   = (float*)d_ws;                       // B*T*H
    float* KE   = KS   + (size_t)B * T * H;           // B*T*H
    float* Msem = KE   + (size_t)B * T * H;           // B*H*H
    float* Mepi = Msem + (size_t)B * H * H;           // B*H*H
    float* rg   = Mepi + (size_t)B * H * H;           // B*2H

    const float alpha = powf(0.95f, 96.0f / (float)L);
    const float scale = 1.0f - alpha;

    // dynamic LDS for the scan kernel: ~195.4 KB (fits 320 KB/WGP)
    const size_t smem_bytes =
        (size_t)(2 * 16 * 512 + 2 * 16 * 516 + 2 * 16 * 516 + 2 * 256 + 32 + 32 + 32) * sizeof(float);
    (void)hipFuncSetAttribute(reinterpret_cast<const void*>(scan_kernel),
                              hipFuncAttributeMaxDynamicSharedMemorySize, (int)smem_bytes);

    dim3 gp(4, 32, B);
    proj_gemm<<<gp, 256, 0, stream>>>(x, W_sem, KS, T, L);
    proj_gemm<<<gp, 256, 0, stream>>>(x, W_epi, KE, T, L);
    scan_kernel<<<B, 1024, smem_bytes, stream>>>(KS, KE, Msem, Mepi, scale, T, L);
    finalize_qr<<<B, 512, 0, stream>>>(x, W_sem, W_epi, null_w, null_b, Msem, Mepi, rg);
    out_gemm<<<8, 256, 0, stream>>>(rg, out_w, out_b, out);
}